// Block_18167711662583
// MI455X (gfx1250) — compile-verified
//
#include <hip/hip_runtime.h>
#include <hip/hip_bf16.h>
#include <math.h>

#define DIM      1024
#define HEADS    16
#define HEAD_DIM 64
#define HIDDEN   4096
#define BATCH    8
#define SEQ      1024
#define MTOT     (BATCH * SEQ)

typedef __attribute__((ext_vector_type(2))) float v2f;
typedef __attribute__((ext_vector_type(8))) float v8f;

__device__ __forceinline__ v8f wmma_f32(v2f a, v2f b, v8f c) {
    // D = A(16x4 f32) * B(4x16 f32) + C(16x16 f32), full fp32 precision
    return __builtin_amdgcn_wmma_f32_16x16x4_f32(
        /*neg_a=*/false, a, /*neg_b=*/false, b,
        /*c_mod=*/(short)0, c, /*reuse_a=*/false, /*reuse_b=*/false);
}

// ---------------------------------------------------------------------------
// Generic WMMA GEMM:
//   C = scale * (A @ op(B)) [+ bias] [GELU] [+ res]
// TRANSB=true : B stored [Nn, K] row-major, computes A@B^T
// TRANSB=false: B stored [K, Nn] row-major, computes A@B
// Block: 256 threads = 8 waves; block tile 256(M) x 64(N).
// Each wave owns a 32x64 register tile: 2 M-subtiles x 4 N-subtiles,
// 8 v8f accumulators. Software-pipelined K loop: fragments for step k+4 are
// loaded into a second register set before the 8 WMMAs of step k issue, so
// vector loads fly under the WMMA burst instead of stalling it.
// blockIdx.z batches over (b, h) with independent strides.
// ---------------------------------------------------------------------------
template <bool TRANSB, bool BIAS, bool RES, bool GELU>
__global__ __launch_bounds__(256) void gemm_wmma(
    const float* __restrict__ A, const float* __restrict__ Bm,
    const float* __restrict__ bias, const float* __restrict__ res,
    float* __restrict__ C,
    int K, int lda, int ldb, int ldc,
    float scale, int hcount,
    long long sAb, long long sAh, long long sBb, long long sBh,
    long long sCb, long long sCh)
{
    const int z  = blockIdx.z;
    const int bb = z / hcount;
    const int hh = z - bb * hcount;
    A  += (long long)bb * sAb + (long long)hh * sAh;
    Bm += (long long)bb * sBb + (long long)hh * sBh;
    C  += (long long)bb * sCb + (long long)hh * sCh;

    const int tid  = threadIdx.x;
    const int lane = tid & 31;
    const int wave = tid >> 5;
    const int m    = lane & 15;   // row-in-tile for A loads / col for B,C
    const int half = lane >> 4;   // upper/lower half of wave

    const int rowBase = blockIdx.y * 256 + wave * 32;  // 32 rows per wave
    const int colBase = blockIdx.x * 64;               // 64 cols per wave

    // A fragment rows for the two M-subtiles
    const float* Arow0 = A + (long long)(rowBase + m) * lda + 2 * half;
    const float* Arow1 = A + (long long)(rowBase + 16 + m) * lda + 2 * half;

    v8f acc[2][4];
#pragma unroll
    for (int i = 0; i < 2; ++i)
#pragma unroll
        for (int j = 0; j < 4; ++j) acc[i][j] = (v8f){};

    if (TRANSB) {
        const float* Br[4];
#pragma unroll
        for (int j = 0; j < 4; ++j)
            Br[j] = Bm + (long long)(colBase + 16 * j + m) * ldb + 2 * half;

        // prologue: fragments for k0 = 0
        v2f a0c = *(const v2f*)(Arow0);
        v2f a1c = *(const v2f*)(Arow1);
        v2f bc[4];
#pragma unroll
        for (int j = 0; j < 4; ++j) bc[j] = *(const v2f*)(Br[j]);

#pragma unroll 2
        for (int k0 = 4; k0 < K; k0 += 4) {
            // issue next-step loads first (independent of current WMMAs)
            v2f a0n = *(const v2f*)(Arow0 + k0);
            v2f a1n = *(const v2f*)(Arow1 + k0);
            v2f bn[4];
#pragma unroll
            for (int j = 0; j < 4; ++j) bn[j] = *(const v2f*)(Br[j] + k0);

#pragma unroll
            for (int j = 0; j < 4; ++j) {
                acc[0][j] = wmma_f32(a0c, bc[j], acc[0][j]);
                acc[1][j] = wmma_f32(a1c, bc[j], acc[1][j]);
            }
            a0c = a0n;
            a1c = a1n;
#pragma unroll
            for (int j = 0; j < 4; ++j) bc[j] = bn[j];
        }
        // epilogue WMMAs for the final K-step
#pragma unroll
        for (int j = 0; j < 4; ++j) {
            acc[0][j] = wmma_f32(a0c, bc[j], acc[0][j]);
            acc[1][j] = wmma_f32(a1c, bc[j], acc[1][j]);
        }
    } else {
        const float* Bp0 = Bm + (long long)(2 * half) * ldb;

        // prologue: fragments for k0 = 0
        v2f a0c = *(const v2f*)(Arow0);
        v2f a1c = *(const v2f*)(Arow1);
        v2f bc[4];
#pragma unroll
        for (int j = 0; j < 4; ++j) {
            const int c = colBase + 16 * j + m;
            bc[j].x = Bp0[c];
            bc[j].y = Bp0[ldb + c];
        }

#pragma unroll 2
        for (int k0 = 4; k0 < K; k0 += 4) {
            v2f a0n = *(const v2f*)(Arow0 + k0);
            v2f a1n = *(const v2f*)(Arow1 + k0);
            const float* Bp = Bp0 + (long long)k0 * ldb;
            v2f bn[4];
#pragma unroll
            for (int j = 0; j < 4; ++j) {
                const int c = colBase + 16 * j + m;
                bn[j].x = Bp[c];
                bn[j].y = Bp[ldb + c];
            }

#pragma unroll
            for (int j = 0; j < 4; ++j) {
                acc[0][j] = wmma_f32(a0c, bc[j], acc[0][j]);
                acc[1][j] = wmma_f32(a1c, bc[j], acc[1][j]);
            }
            a0c = a0n;
            a1c = a1n;
#pragma unroll
            for (int j = 0; j < 4; ++j) bc[j] = bn[j];
        }
#pragma unroll
        for (int j = 0; j < 4; ++j) {
            acc[0][j] = wmma_f32(a0c, bc[j], acc[0][j]);
            acc[1][j] = wmma_f32(a1c, bc[j], acc[1][j]);
        }
    }

    float biasv[4];
    if (BIAS) {
#pragma unroll
        for (int j = 0; j < 4; ++j) biasv[j] = bias[colBase + 16 * j + m];
    }

#pragma unroll
    for (int i = 0; i < 2; ++i) {
#pragma unroll
        for (int r = 0; r < 8; ++r) {
            const int row = rowBase + i * 16 + r + 8 * half;
#pragma unroll
            for (int j = 0; j < 4; ++j) {
                const int col = colBase + 16 * j + m;
                float v = acc[i][j][r] * scale;
                if (BIAS) v += biasv[j];
                if (GELU) v = 0.5f * v * (1.0f + erff(v * 0.70710678118f));
                const long long o = (long long)row * ldc + col;
                if (RES) v += res[o];
                C[o] = v;
            }
        }
    }
}

// ---------------------------------------------------------------------------
// LayerNorm over rows of 1024; one block (256 threads, 4 elems/thread) per row
// ---------------------------------------------------------------------------
__global__ __launch_bounds__(256) void ln_kernel(
    const float* __restrict__ x, const float* __restrict__ g,
    const float* __restrict__ be, float* __restrict__ out)
{
    __shared__ float r1[256];
    __shared__ float r2[256];
    const int row = blockIdx.x;
    const int tid = threadIdx.x;
    const float4 v = ((const float4*)(x + (long long)row * DIM))[tid];
    float s  = v.x + v.y + v.z + v.w;
    float ss = v.x * v.x + v.y * v.y + v.z * v.z + v.w * v.w;
    r1[tid] = s; r2[tid] = ss;
    __syncthreads();
    for (int off = 128; off > 0; off >>= 1) {
        if (tid < off) { r1[tid] += r1[tid + off]; r2[tid] += r2[tid + off]; }
        __syncthreads();
    }
    const float mean = r1[0] * (1.0f / DIM);
    const float var  = r2[0] * (1.0f / DIM) - mean * mean;
    const float inv  = rsqrtf(var + 1e-5f);
    const float4 gg = ((const float4*)g)[tid];
    const float4 bb = ((const float4*)be)[tid];
    float4 o;
    o.x = (v.x - mean) * inv * gg.x + bb.x;
    o.y = (v.y - mean) * inv * gg.y + bb.y;
    o.z = (v.z - mean) * inv * gg.z + bb.z;
    o.w = (v.w - mean) * inv * gg.w + bb.w;
    ((float4*)(out + (long long)row * DIM))[tid] = o;
}

// ---------------------------------------------------------------------------
// In-place softmax over rows of 1024 (the attention scores in d_out)
// ---------------------------------------------------------------------------
__global__ __launch_bounds__(256) void softmax_kernel(float* __restrict__ attn)
{
    __shared__ float red[256];
    const long long row = blockIdx.x;
    float* p = attn + row * (long long)SEQ;
    const int tid = threadIdx.x;
    float4 v = ((float4*)p)[tid];

    float mx = fmaxf(fmaxf(v.x, v.y), fmaxf(v.z, v.w));
    red[tid] = mx;
    __syncthreads();
    for (int off = 128; off > 0; off >>= 1) {
        if (tid < off) red[tid] = fmaxf(red[tid], red[tid + off]);
        __syncthreads();
    }
    mx = red[0];
    __syncthreads();

    v.x = __expf(v.x - mx);
    v.y = __expf(v.y - mx);
    v.z = __expf(v.z - mx);
    v.w = __expf(v.w - mx);
    red[tid] = v.x + v.y + v.z + v.w;
    __syncthreads();
    for (int off = 128; off > 0; off >>= 1) {
        if (tid < off) red[tid] += red[tid + off];
        __syncthreads();
    }
    const float inv = 1.0f / red[0];
    v.x *= inv; v.y *= inv; v.z *= inv; v.w *= inv;
    ((float4*)p)[tid] = v;
}

// ---------------------------------------------------------------------------
extern "C" void kernel_launch(void* const* d_in, const int* in_sizes, int n_in,
                              void* d_out, int out_size, void* d_ws, size_t ws_size,
                              hipStream_t stream)
{
    const float* x     = (const float*)d_in[0];
    const float* ln1g  = (const float*)d_in[1];
    const float* ln1b  = (const float*)d_in[2];
    const float* wqkv  = (const float*)d_in[3];
    const float* wproj = (const float*)d_in[4];
    const float* bproj = (const float*)d_in[5];
    const float* ln2g  = (const float*)d_in[6];
    const float* ln2b  = (const float*)d_in[7];
    const float* wfc1  = (const float*)d_in[8];
    const float* bfc1  = (const float*)d_in[9];
    const float* wfc2  = (const float*)d_in[10];
    const float* bfc2  = (const float*)d_in[11];

    float* outx = (float*)d_out;                       // [8192, 1024]
    float* attn = outx + (long long)MTOT * DIM;        // [B*H, 1024, 1024]

    float* xn  = (float*)d_ws;                         // [8192, 1024]
    float* qkv = xn  + (long long)MTOT * DIM;          // [8192, 3072]
    float* y   = qkv + (long long)MTOT * 3 * DIM;      // [8192, 1024]
    float* hbf = y   + (long long)MTOT * DIM;          // [8192, 4096]

    const dim3 blk(256);

    // 1) xn = LN1(x)
    ln_kernel<<<MTOT, blk, 0, stream>>>(x, ln1g, ln1b, xn);

    // 2) qkv = xn @ w_qkv^T   [8192 x 3072]
    gemm_wmma<true, false, false, false><<<dim3(3 * DIM / 64, MTOT / 256, 1), blk, 0, stream>>>(
        xn, wqkv, nullptr, nullptr, qkv,
        DIM, DIM, DIM, 3 * DIM, 1.0f, 1, 0, 0, 0, 0, 0, 0);

    // 3) scores = scale * q @ k^T  -> attn region of d_out
    gemm_wmma<true, false, false, false><<<dim3(SEQ / 64, SEQ / 256, BATCH * HEADS), blk, 0, stream>>>(
        qkv /*q*/, qkv + DIM /*k*/, nullptr, nullptr, attn,
        HEAD_DIM, 3 * DIM, 3 * DIM, SEQ, 0.125f, HEADS,
        (long long)SEQ * 3 * DIM, (long long)HEAD_DIM,          // A (b,h) strides
        (long long)SEQ * 3 * DIM, (long long)HEAD_DIM,          // B (b,h) strides
        (long long)HEADS * SEQ * SEQ, (long long)SEQ * SEQ);    // C (b,h) strides

    // 4) softmax rows, in place
    softmax_kernel<<<BATCH * HEADS * SEQ, blk, 0, stream>>>(attn);

    // 5) y = attn @ v   (NN GEMM, write heads into [B*N, 1024] layout)
    gemm_wmma<false, false, false, false><<<dim3(HEAD_DIM / 64, SEQ / 256, BATCH * HEADS), blk, 0, stream>>>(
        attn, qkv + 2 * DIM /*v*/, nullptr, nullptr, y,
        SEQ, SEQ, 3 * DIM, DIM, 1.0f, HEADS,
        (long long)HEADS * SEQ * SEQ, (long long)SEQ * SEQ,
        (long long)SEQ * 3 * DIM, (long long)HEAD_DIM,
        (long long)SEQ * DIM, (long long)HEAD_DIM);

    // 6) x1 = x + y @ w_proj^T + b_proj  -> outx
    gemm_wmma<true, true, true, false><<<dim3(DIM / 64, MTOT / 256, 1), blk, 0, stream>>>(
        y, wproj, bproj, x, outx,
        DIM, DIM, DIM, DIM, 1.0f, 1, 0, 0, 0, 0, 0, 0);

    // 7) xn = LN2(x1)
    ln_kernel<<<MTOT, blk, 0, stream>>>(outx, ln2g, ln2b, xn);

    // 8) h = gelu(xn @ w_fc1^T + b_fc1)
    gemm_wmma<true, true, false, true><<<dim3(HIDDEN / 64, MTOT / 256, 1), blk, 0, stream>>>(
        xn, wfc1, bfc1, nullptr, hbf,
        DIM, DIM, DIM, HIDDEN, 1.0f, 1, 0, 0, 0, 0, 0, 0);

    // 9) out = x1 + h @ w_fc2^T + b_fc2  (in-place residual on outx)
    gemm_wmma<true, true, true, false><<<dim3(DIM / 64, MTOT / 256, 1), blk, 0, stream>>>(
        hbf, wfc2, bfc2, outx, outx,
        HIDDEN, HIDDEN, HIDDEN, DIM, 1.0f, 1, 0, 0, 0, 0, 0, 0);
}